// Net_80539226734613
// MI455X (gfx1250) — compile-verified
//
#include <hip/hip_runtime.h>
#include <cstdint>
#include <cstddef>

// ---------------------------------------------------------------------------
// CDNA5 (gfx1250) DeepPoly bound propagation.
// Heavy GEMMs use V_WMMA_F32_16X16X4_F32 (fp32 to match reference numerics).
// B tiles are moved with global_load_async_to_lds_b128 when available.
// ---------------------------------------------------------------------------

typedef float v2f __attribute__((ext_vector_type(2)));
typedef float v8f __attribute__((ext_vector_type(8)));
typedef int   v4i __attribute__((ext_vector_type(4)));

#define TM 128
#define TN 128
#define TK 16

#ifdef __HIP_DEVICE_COMPILE__
#if __has_builtin(__builtin_amdgcn_global_load_async_to_lds_b128) && \
    __has_builtin(__builtin_amdgcn_s_wait_asynccnt)
#define HAVE_ASYNC_LDS 1
#endif
#endif

// Address-space qualified element types for the async-LDS builtin:
// param0 is 'v4i __device__*' (AS1), param1 is the LDS-side pointer (AS3).
typedef v4i __attribute__((address_space(1))) g_v4i;
typedef v4i __attribute__((address_space(3))) l_v4i;

// ---------------------------------------------------------------------------
// GEMM: C[M,N] = scale(A[M,K]) @ B[K,N]
//   scale(a)[r,k] = max(a,0)*cw[k] + min(a,0)*dw[k]
// K must be a multiple of TK (always K == h == 2048 here). N % 4 == 0.
// ---------------------------------------------------------------------------
__global__ __launch_bounds__(256) void k_gemm_scaled(
    const float* __restrict__ A, const float* __restrict__ B,
    float* __restrict__ C, int M, int N, int K,
    const float* __restrict__ cw, const float* __restrict__ dw)
{
  __shared__ float sA[TK][TM + 4];   // transposed: sA[k][m]
  __shared__ float sB[TK][TN];       // natural:    sB[k][n]

  const int tid  = threadIdx.x;
  const int lane = tid & 31;
  const int wave = tid >> 5;
  const int hi   = lane >> 4;   // half-wave select
  const int l16  = lane & 15;
  const int wm   = (wave >> 2) * 64;   // 2 waves along M
  const int wn   = (wave & 3) * 32;    // 4 waves along N

  const int bm = blockIdx.y * TM;
  const int bn = blockIdx.x * TN;
  const bool nfull = (bn + TN <= N);

  v8f acc[4][2];
#pragma unroll
  for (int mi = 0; mi < 4; ++mi)
#pragma unroll
    for (int ni = 0; ni < 2; ++ni)
#pragma unroll
      for (int e = 0; e < 8; ++e) acc[mi][ni][e] = 0.0f;

  for (int k0 = 0; k0 < K; k0 += TK) {
    // ---- A tile: load, sign-scale, transpose into LDS -------------------
#pragma unroll
    for (int it = 0; it < 2; ++it) {
      int id  = tid + it * 256;      // 512 float4 slots
      int row = id >> 2;             // 0..127
      int kg  = (id & 3) * 4;        // 0,4,8,12
      int gr  = bm + row;
      int gk  = k0 + kg;
      if (gr < M) {
        const float4 a = *(const float4*)(A + (size_t)gr * K + gk);
        float av[4] = {a.x, a.y, a.z, a.w};
#pragma unroll
        for (int e = 0; e < 4; ++e) {
          float x  = av[e];
          float sv = fmaxf(x, 0.0f) * cw[gk + e] + fminf(x, 0.0f) * dw[gk + e];
          sA[kg + e][row] = sv;
        }
      } else {
#pragma unroll
        for (int e = 0; e < 4; ++e) sA[kg + e][row] = 0.0f;
      }
    }

    // ---- B tile ---------------------------------------------------------
#ifdef HAVE_ASYNC_LDS
    if (nfull) {
#pragma unroll
      for (int it = 0; it < 2; ++it) {
        int id = tid + it * 256;
        int k  = id >> 5;            // 0..15
        int ng = (id & 31) * 4;      // 0..124
        const float* gp = B + (size_t)(k0 + k) * N + bn + ng;
        __builtin_amdgcn_global_load_async_to_lds_b128(
            (g_v4i*)(void*)gp, (l_v4i*)(void*)&sB[k][ng], 0, 0);
      }
      __builtin_amdgcn_s_wait_asynccnt(0);
    } else
#endif
    {
#pragma unroll
      for (int it = 0; it < 2; ++it) {
        int id = tid + it * 256;
        int k  = id >> 5;
        int ng = (id & 31) * 4;
        int gc = bn + ng;
        float4 bv = make_float4(0.f, 0.f, 0.f, 0.f);
        if (gc < N)  // N % 4 == 0 and gc % 4 == 0 -> whole float4 in range
          bv = *(const float4*)(B + (size_t)(k0 + k) * N + gc);
        *(float4*)&sB[k][ng] = bv;
      }
    }
    __syncthreads();

    // ---- WMMA: 4 k-steps of 4 per tile ---------------------------------
#pragma unroll
    for (int kk = 0; kk < TK; kk += 4) {
      v2f af[4], bf[2];
#pragma unroll
      for (int mi = 0; mi < 4; ++mi) {
        int r = wm + mi * 16 + l16;
        af[mi].x = sA[kk + 2 * hi + 0][r];   // A 16x4: K = kk + 2*hi + v
        af[mi].y = sA[kk + 2 * hi + 1][r];
      }
#pragma unroll
      for (int ni = 0; ni < 2; ++ni) {
        int c = wn + ni * 16 + l16;
        bf[ni].x = sB[kk + 2 * hi + 0][c];
        bf[ni].y = sB[kk + 2 * hi + 1][c];
      }
#pragma unroll
      for (int mi = 0; mi < 4; ++mi)
#pragma unroll
        for (int ni = 0; ni < 2; ++ni)
          acc[mi][ni] = __builtin_amdgcn_wmma_f32_16x16x4_f32(
              false, af[mi], false, bf[ni], (short)0, acc[mi][ni], false, false);
    }
    __syncthreads();
  }

  // ---- store: C/D layout M = e + 8*hi, N = l16 --------------------------
#pragma unroll
  for (int mi = 0; mi < 4; ++mi)
#pragma unroll
    for (int ni = 0; ni < 2; ++ni)
#pragma unroll
      for (int e = 0; e < 8; ++e) {
        int r = bm + wm + mi * 16 + hi * 8 + e;
        int c = bn + wn + ni * 16 + l16;
        if (r < M && c < N) C[(size_t)r * N + c] = acc[mi][ni][e];
      }
}

// ---------------------------------------------------------------------------
// out[r] = sum_k max(A[r,k],0)*p[k] + min(A[r,k],0)*q[k]  (+ addb[r])
// ---------------------------------------------------------------------------
__global__ __launch_bounds__(256) void k_signed_reduce(
    const float* __restrict__ A, int M, int K,
    const float* __restrict__ p, const float* __restrict__ q,
    const float* __restrict__ addb, float* __restrict__ out)
{
  int row = blockIdx.x;
  if (row >= M) return;
  const float* a = A + (size_t)row * K;
  float s = 0.0f;
  for (int k = threadIdx.x; k < K; k += 256) {
    float v = a[k];
    s += fmaxf(v, 0.0f) * p[k] + fminf(v, 0.0f) * q[k];
  }
  __shared__ float red[256];
  red[threadIdx.x] = s;
  __syncthreads();
  for (int st = 128; st > 0; st >>= 1) {
    if (threadIdx.x < st) red[threadIdx.x] += red[threadIdx.x + st];
    __syncthreads();
  }
  if (threadIdx.x == 0) out[row] = red[0] + (addb ? addb[row] : 0.0f);
}

// ---------------------------------------------------------------------------
// SPU relaxation parameters
// ---------------------------------------------------------------------------
__device__ __forceinline__ float spu_f(float x) {
  float s = 1.0f / (1.0f + expf(x));      // sigmoid(-x)
  return (x >= 0.0f) ? (x * x - 0.5f) : (s - 1.0f);
}
__device__ __forceinline__ float dspu_f(float x) {
  float s = 1.0f / (1.0f + expf(x));
  return (x >= 0.0f) ? (2.0f * x) : (-s * (1.0f - s));
}

__global__ void k_spu_params(const float* __restrict__ lf, const float* __restrict__ uf,
                             const float* __restrict__ pl, const float* __restrict__ pu,
                             float* __restrict__ wl, float* __restrict__ wu,
                             float* __restrict__ bl, float* __restrict__ bu, int n)
{
  int i = blockIdx.x * blockDim.x + threadIdx.x;
  if (i >= n) return;
  float l = lf[i], u = uf[i];
  float k1 = dspu_f(l), k2 = dspu_f(u);
  float klo = fminf(k1, k2), khi = fmaxf(k1, k2);
  float cl = fminf(fmaxf(pl[i], klo), khi);
  float cu = fminf(fmaxf(pu[i], klo), khi);
  float pts[4] = {l, u, 0.5f * (l + u), fminf(fmaxf(0.0f, l), u)};
  float bmin = 3.0e38f, bmax = -3.0e38f;
#pragma unroll
  for (int t = 0; t < 4; ++t) {
    float sp = spu_f(pts[t]);
    bmin = fminf(bmin, sp - cl * pts[t]);
    bmax = fmaxf(bmax, sp - cu * pts[t]);
  }
  wl[i] = cl; wu[i] = cu; bl[i] = bmin; bu[i] = bmax;
}

// P[i] = w[i]*b[i] + bb[i]
__global__ void k_make_pq(const float* __restrict__ w, const float* __restrict__ bb,
                          const float* __restrict__ b, float* __restrict__ P, int n)
{
  int i = blockIdx.x * blockDim.x + threadIdx.x;
  if (i < n) P[i] = w[i] * b[i] + bb[i];
}

// Build padded (16 x h) output-specification matrix and its bias.
__global__ void k_build_final(const float* __restrict__ W4, const float* __restrict__ b4,
                              const int* __restrict__ tl, float* __restrict__ v,
                              float* __restrict__ vb, int h, int nout)
{
  int t = *tl;
  int rows = nout - 1;
  int i = blockIdx.x * blockDim.x + threadIdx.x;
  if (i < 16 * h) {
    int r = i / h, c = i % h;
    float val = 0.0f;
    if (r < rows) {
      int cls = (r < t) ? r : r + 1;
      val = W4[(size_t)cls * h + c] - W4[(size_t)t * h + c];
    }
    v[i] = val;
  }
  if (i < 16) {
    float bv = 0.0f;
    if (i < rows) {
      int cls = (i < t) ? i : i + 1;
      bv = b4[cls] - b4[t];
    }
    vb[i] = bv;
  }
}

// ---------------------------------------------------------------------------
extern "C" void kernel_launch(void* const* d_in, const int* in_sizes, int n_in,
                              void* d_out, int out_size, void* d_ws, size_t ws_size,
                              hipStream_t stream) {
  (void)n_in; (void)out_size; (void)ws_size;
  const float* W1 = (const float*)d_in[0];
  const float* b1 = (const float*)d_in[1];
  const float* W2 = (const float*)d_in[2];
  const float* b2 = (const float*)d_in[3];
  const float* W3 = (const float*)d_in[4];
  const float* b3 = (const float*)d_in[5];
  const float* W4 = (const float*)d_in[6];
  const float* b4 = (const float*)d_in[7];
  const float* l_in = (const float*)d_in[8];
  const float* u_in = (const float*)d_in[9];
  const float* pl1 = (const float*)d_in[10];
  const float* pu1 = (const float*)d_in[11];
  const float* pl2 = (const float*)d_in[12];
  const float* pu2 = (const float*)d_in[13];
  const float* pl3 = (const float*)d_in[14];
  const float* pu3 = (const float*)d_in[15];
  const int*   tl  = (const int*)d_in[16];

  const int h    = in_sizes[1];   // 2048
  const int din  = in_sizes[8];   // 784
  const int nout = in_sizes[7];   // 10

  float* W = (float*)d_ws;
  size_t o = 0;
  auto A_ = [&](size_t n) { float* r = W + o; o += n; return r; };

  float *lf1 = A_(h), *uf1 = A_(h), *wl1 = A_(h), *wu1 = A_(h), *bl1 = A_(h), *bu1 = A_(h);
  float *lf2 = A_(h), *uf2 = A_(h), *wl2 = A_(h), *wu2 = A_(h), *bl2 = A_(h), *bu2 = A_(h);
  float *lf3 = A_(h), *uf3 = A_(h), *wl3 = A_(h), *wu3 = A_(h), *bl3 = A_(h), *bu3 = A_(h);
  float *P1 = A_(h), *Q1 = A_(h), *P2 = A_(h), *Q2 = A_(h), *P3 = A_(h), *Q3 = A_(h);
  float *betal = A_(h), *betau = A_(h);
  float *gl = A_(h), *gu = A_(h), *g2l = A_(h), *g2u = A_(h);
  float *Tl = A_((size_t)h * din), *Tu = A_((size_t)h * din);   // reused as El/Eu
  float *Cl = A_((size_t)h * h),   *Cu = A_((size_t)h * h);
  float *v  = A_((size_t)16 * h);
  float *F1 = A_((size_t)16 * h), *F2 = A_((size_t)16 * h), *F3 = A_((size_t)16 * din);
  float *h0 = A_(16), *h1 = A_(16), *h2 = A_(16), *vb = A_(16);

  auto gemm = [&](const float* A, const float* B, float* C, int M, int N, int K,
                  const float* cw, const float* dw) {
    dim3 g((N + TN - 1) / TN, (M + TM - 1) / TM);
    k_gemm_scaled<<<g, 256, 0, stream>>>(A, B, C, M, N, K, cw, dw);
  };
  auto sred = [&](const float* A, int M, int K, const float* p, const float* q,
                  const float* ab, float* out) {
    k_signed_reduce<<<M, 256, 0, stream>>>(A, M, K, p, q, ab, out);
  };
  const int EB = (h + 255) / 256;

  // ---- Stage 1: interval bound through W1 -------------------------------
  sred(W1, h, din, l_in, u_in, b1, lf1);
  sred(W1, h, din, u_in, l_in, b1, uf1);
  k_spu_params<<<EB, 256, 0, stream>>>(lf1, uf1, pl1, pu1, wl1, wu1, bl1, bu1, h);
  k_make_pq<<<EB, 256, 0, stream>>>(wl1, bl1, b1, P1, h);
  k_make_pq<<<EB, 256, 0, stream>>>(wu1, bu1, b1, Q1, h);

  // ---- Stage 2 ----------------------------------------------------------
  sred(W2, h, h, P1, Q1, b2, betal);
  sred(W2, h, h, Q1, P1, b2, betau);
  gemm(W2, W1, Tl, h, din, h, wl1, wu1);
  gemm(W2, W1, Tu, h, din, h, wu1, wl1);
  sred(Tl, h, din, l_in, u_in, betal, lf2);
  sred(Tu, h, din, u_in, l_in, betau, uf2);
  k_spu_params<<<EB, 256, 0, stream>>>(lf2, uf2, pl2, pu2, wl2, wu2, bl2, bu2, h);
  k_make_pq<<<EB, 256, 0, stream>>>(wl2, bl2, b2, P2, h);
  k_make_pq<<<EB, 256, 0, stream>>>(wu2, bu2, b2, Q2, h);

  // ---- Stage 3 ----------------------------------------------------------
  sred(W3, h, h, P2, Q2, b3, gl);
  sred(W3, h, h, Q2, P2, b3, gu);
  gemm(W3, W2, Cl, h, h, h, wl2, wu2);
  gemm(W3, W2, Cu, h, h, h, wu2, wl2);
  sred(Cl, h, h, P1, Q1, gl, g2l);
  sred(Cu, h, h, Q1, P1, gu, g2u);
  gemm(Cl, W1, Tl, h, din, h, wl1, wu1);
  gemm(Cu, W1, Tu, h, din, h, wu1, wl1);
  sred(Tl, h, din, l_in, u_in, g2l, lf3);
  sred(Tu, h, din, u_in, l_in, g2u, uf3);
  k_spu_params<<<EB, 256, 0, stream>>>(lf3, uf3, pl3, pu3, wl3, wu3, bl3, bu3, h);
  k_make_pq<<<EB, 256, 0, stream>>>(wl3, bl3, b3, P3, h);
  k_make_pq<<<EB, 256, 0, stream>>>(wu3, bu3, b3, Q3, h);

  // ---- Final back-substitution (upper bound only) -----------------------
  k_build_final<<<(16 * h + 255) / 256, 256, 0, stream>>>(W4, b4, tl, v, vb, h, nout);
  sred(v, 16, h, Q3, P3, vb, h0);
  gemm(v, W3, F1, 16, h, h, wu3, wl3);
  sred(F1, 16, h, Q2, P2, h0, h1);
  gemm(F1, W2, F2, 16, h, h, wu2, wl2);
  sred(F2, 16, h, Q1, P1, h1, h2);
  gemm(F2, W1, F3, 16, din, h, wu1, wl1);
  sred(F3, nout - 1, din, u_in, l_in, h2, (float*)d_out);
}